// CapsuleMaxPooling_48515950576312
// MI455X (gfx1250) — compile-verified
//
#include <hip/hip_runtime.h>
#include <hip/hip_bf16.h>
#include <stdint.h>

// Capsule max pooling, MI455X (gfx1250).
// inp: [B=32, C=32, H=64, W=64, D=8] f32 ; k=2 ; out: [32,32,32,32,8] f32.
// Memory-bound streaming op (~160 MiB total traffic -> ~7us at 23.3 TB/s).
// CDNA5 path: async global->LDS staging (ASYNCcnt) + wave32 tiling + NT stores.

#define B_  32
#define C_  32
#define H_  64
#define W_  64
#define D_  8
#define HK_ 32
#define WK_ 32

typedef __attribute__((ext_vector_type(4))) float v4f;  // native vector (NT-store legal)

// One block = one (b,c) image x 8 output rows (= 16 contiguous input rows = 32 KB).
__global__ __launch_bounds__(256) void capsule_maxpool_kernel(
    const float* __restrict__ inp, float* __restrict__ out) {
  __shared__ __align__(16) v4f smem[2048];  // 32 KB staging buffer

  const int t   = threadIdx.x;          // 0..255
  const int bid = blockIdx.x;           // 0..4095
  const int bc  = bid >> 2;             // image index b*C + c (0..1023)
  const int hg  = bid & 3;              // group of 8 output rows (0..3)

  // ---- async stage: 32 KB contiguous global -> LDS -------------------------
  // Input rows h = 16*hg .. 16*hg+15 of image (b,c) are one flat 32 KB chunk.
  const float* gbase =
      inp + (size_t)bc * (H_ * W_ * D_) + (size_t)hg * 16 * (W_ * D_);
  // Flat pointer into LDS aperture: low 32 bits are the LDS byte offset.
  const uint32_t lds0 = (uint32_t)(uintptr_t)smem;

#pragma unroll
  for (int j = 0; j < 8; ++j) {
    const int chunk = j * 256 + t;                       // v4f index 0..2047
    const uint64_t ga = (uint64_t)(uintptr_t)(gbase + chunk * 4);
    const uint32_t la = lds0 + (uint32_t)chunk * 16u;
    asm volatile("global_load_async_to_lds_b128 %0, %1, off"
                 :
                 : "v"(la), "v"(ga)
                 : "memory");
  }
  // Each wave drains its own ASYNCcnt, then the barrier makes all staging
  // visible workgroup-wide.
  asm volatile("s_wait_asynccnt 0" ::: "memory");
  __syncthreads();

  // ---- compute: one output capsule per thread ------------------------------
  const int hk = t >> 5;                 // local output row   (0..7)
  const int wk = t & 31;                 // output column      (0..31)
  const int r0 = hk * 2;                 // input row in LDS   (0..14, even)
  const int w0 = wk * 2;                 // input col          (0..62, even)

  // LDS layout: capsule (r, w) -> v4f pair at index (r*64 + w)*2.
  const v4f* s = smem;
  const v4f a0 = s[(r0 * 64 + w0) * 2 + 0],       a1 = s[(r0 * 64 + w0) * 2 + 1];
  const v4f b0 = s[(r0 * 64 + w0 + 1) * 2 + 0],   b1 = s[(r0 * 64 + w0 + 1) * 2 + 1];
  const v4f c0 = s[((r0 + 1) * 64 + w0) * 2 + 0], c1 = s[((r0 + 1) * 64 + w0) * 2 + 1];
  const v4f d0 = s[((r0 + 1) * 64 + w0 + 1) * 2 + 0],
            d1 = s[((r0 + 1) * 64 + w0 + 1) * 2 + 1];

  auto nrm = [](v4f x, v4f y) {
    return x.x * x.x + x.y * x.y + x.z * x.z + x.w * x.w +
           y.x * y.x + y.y * y.y + y.z * y.z + y.w * y.w;
  };
  const float n0 = nrm(a0, a1);
  const float n1 = nrm(b0, b1);
  const float n2 = nrm(c0, c1);
  const float n3 = nrm(d0, d1);

  // First-max semantics (strict > so earlier window index wins ties).
  // Window order matches reference flatten: (dh,dw) = (0,0),(0,1),(1,0),(1,1).
  float best = n0;
  v4f   o0 = a0, o1 = a1;
  if (n1 > best) { best = n1; o0 = b0; o1 = b1; }
  if (n2 > best) { best = n2; o0 = c0; o1 = c1; }
  if (n3 > best) { best = n3; o0 = d0; o1 = d1; }

  // Output rows hk0..hk0+7 of image (b,c) are contiguous: 8 KB per block.
  v4f* ob = (v4f*)(out + (size_t)bc * (HK_ * WK_ * D_) +
                   (size_t)hg * (8 * WK_ * D_)) +
            (size_t)t * 2;
  __builtin_nontemporal_store(o0, ob);
  __builtin_nontemporal_store(o1, ob + 1);
}

extern "C" void kernel_launch(void* const* d_in, const int* in_sizes, int n_in,
                              void* d_out, int out_size, void* d_ws, size_t ws_size,
                              hipStream_t stream) {
  (void)in_sizes; (void)n_in; (void)d_ws; (void)ws_size; (void)out_size;
  const float* inp = (const float*)d_in[0];
  // d_in[1] is kernel_size (==2); geometry is compile-time specialized.
  float* out = (float*)d_out;

  const int blocks = B_ * C_ * (HK_ / 8);  // 4096
  capsule_maxpool_kernel<<<blocks, 256, 0, stream>>>(inp, out);
}